// BrainGAT_68693706932795
// MI455X (gfx1250) — compile-verified
//
#include <hip/hip_runtime.h>

#define HEADS 4
#define NEG_SLOPE 0.2f

typedef __attribute__((ext_vector_type(2))) float v2f;
typedef __attribute__((ext_vector_type(8))) float v8f;
typedef __attribute__((ext_vector_type(4))) unsigned int v4u;
typedef __attribute__((ext_vector_type(8))) int v8i;
typedef __attribute__((ext_vector_type(4))) int v4i;

// ---------------------------------------------------------------------------
// Generic fill (init segment-max keys, segment-sums, accumulators)
// ---------------------------------------------------------------------------
__global__ void fill_u32_kernel(unsigned int* __restrict__ p, unsigned int v, int n) {
    int t = blockIdx.x * blockDim.x + threadIdx.x;
    if (t < n) p[t] = v;
}

// ---------------------------------------------------------------------------
// WMMA fp32 GEMM with TDM-staged A tile:
//   Out[M,Nout] = X[M,K] @ W[Nout,K]^T + bias[Nout]
//
// Block = 128 threads (4 waves). Block computes a 16(M) x 64(N) strip.
// The 16xK X-strip is DMA'd once into LDS by the Tensor Data Mover
// (tensor_load_to_lds, D# built per CDNA5 ISA ch.8), with pad_amount=2 dwords
// every K dwords -> LDS row stride K+2 (bank-conflict-free A reads).
// Each wave owns one 16x16 tile: A-fragments from LDS (ds_load_b64),
// B-fragments from global (W rows are L1/L2 resident), accumulation via
// V_WMMA_F32_16X16X4_F32.
//
// A-frag (16x4 MxK): lanes 0-15 rows M=0..15 hold K+0/1; lanes 16-31 K+2/3.
// C/D (16x16 f32): VGPR j -> row (j + 8*half), col = lane&15.
// Requires M%16==0, Nout%64==0, K a power of two >= 4 (holds: 128/256).
// ---------------------------------------------------------------------------
__global__ void wmma_gemm_bias(const float* __restrict__ X,
                               const float* __restrict__ W,
                               const float* __restrict__ bias,
                               float* __restrict__ Out,
                               int K, int Nout)
{
    __shared__ float tileX[16 * 258];      // 16 rows, stride K+2 (max 258)

    const int lane = threadIdx.x & 31;
    const int wv   = threadIdx.x >> 5;     // 0..3
    const int l    = lane & 15;
    const int half = lane >> 4;            // 0: K+0/1   1: K+2/3
    const int m0   = blockIdx.x * 16;
    const int n0   = blockIdx.y * 64 + wv * 16;
    const int RS   = K + 2;                // padded LDS row stride (dwords)

    // ---- TDM: DMA 16 x K fp32 strip X[m0 .. m0+15][0..K) into LDS ----
    if (wv == 0) {
        unsigned long long ga = (unsigned long long)(const void*)(X + (size_t)m0 * K);
        unsigned int ldsOff   = (unsigned int)(unsigned long long)(const void*)tileX;
        // pad_interval code: pad after 2^(code+1) dwords == K dwords
        unsigned int icode = (unsigned int)(31 - __builtin_clz((unsigned)K)) - 1u;

        v4u g0;
        g0[0] = 1u;                                        // count=1 (valid D#)
        g0[1] = ldsOff;                                    // lds_addr (bytes)
        g0[2] = (unsigned int)ga;                          // global_addr[31:0]
        g0[3] = ((unsigned int)(ga >> 32) & 0x01FFFFFFu)   // global_addr[56:32]
                | 0x80000000u;                             // type=2 ("image")

        v8i g1;
        g1[0] = (int)(0x00020000u                          // data_size=2 (4B)
                      | (1u << 20)                         // pad_enable
                      | (icode << 22)                      // pad_interval
                      | (1u << 25));                       // pad_amount=2 dwords
        g1[1] = (int)((unsigned)K << 16);                  // tensor_dim0[15:0]
        g1[2] = (int)(16u << 16);                          // tensor_dim1 = 16
        g1[3] = (int)((unsigned)K << 16);                  // tile_dim0 = K
        g1[4] = 16;                                        // tile_dim1 = 16
        g1[5] = K;                                         // tensor_dim0_stride
        g1[6] = 0;
        g1[7] = 0;

        v4i g2 = {0, 0, 0, 0};
        v4i g3 = {0, 0, 0, 0};
        v8i g4 = {0, 0, 0, 0, 0, 0, 0, 0};
        __builtin_amdgcn_tensor_load_to_lds(g0, g1, g2, g3, g4, 0);
        __builtin_amdgcn_s_wait_tensorcnt(0);
    }
    __syncthreads();

    // ---- WMMA main loop: A from LDS, B from global ----
    const float* arow = tileX + l * RS + 2 * half;
    const float* wrow = W + (size_t)(n0 + l) * K + 2 * half;

    v8f acc = {};
    for (int k = 0; k < K; k += 4) {
        v2f a = *(const v2f*)(arow + k);
        v2f b = *(const v2f*)(wrow + k);
        acc = __builtin_amdgcn_wmma_f32_16x16x4_f32(
            /*neg_a=*/false, a, /*neg_b=*/false, b,
            /*c_mod=*/(short)0, acc, /*reuse_a=*/false, /*reuse_b=*/false);
    }

    const float bv = bias[n0 + l];
    const int   rb = m0 + half * 8;
#pragma unroll
    for (int j = 0; j < 8; ++j)
        Out[(size_t)(rb + j) * Nout + (n0 + l)] = acc[j] + bv;
}

// ---------------------------------------------------------------------------
// Wave-wide sum reduction (wave32)
// ---------------------------------------------------------------------------
__device__ __forceinline__ float wave_reduce_sum(float v) {
    v += __shfl_xor(v, 16, 32);
    v += __shfl_xor(v,  8, 32);
    v += __shfl_xor(v,  4, 32);
    v += __shfl_xor(v,  2, 32);
    v += __shfl_xor(v,  1, 32);
    return v;
}

// monotone uint encoding of float for atomicMax-based segment max
__device__ __forceinline__ unsigned int f32_ordered_key(float f) {
    unsigned int b = __float_as_uint(f);
    return (b & 0x80000000u) ? ~b : (b | 0x80000000u);
}
__device__ __forceinline__ float f32_from_key(unsigned int k) {
    unsigned int b = (k & 0x80000000u) ? (k & 0x7FFFFFFFu) : ~k;
    return __uint_as_float(b);
}

// ---------------------------------------------------------------------------
// Pass 1: per-edge GATv2 logits + segment max.  Wave per edge.
// logit[e,h] = att[h] . leaky_relu(xl[src] + xr[dst])
// ---------------------------------------------------------------------------
__global__ void edge_logits_max(const long long* __restrict__ ei, int E, int Etot,
                                const float* __restrict__ xl,
                                const float* __restrict__ xr,
                                const float* __restrict__ att,   // [HEADS, CH]
                                float* __restrict__ logits,      // [Etot, HEADS]
                                unsigned int* __restrict__ mkey, // [N, HEADS]
                                int CH)
{
    const int wave = (blockIdx.x * blockDim.x + threadIdx.x) >> 5;
    const int lane = threadIdx.x & 31;
    if (wave >= Etot) return;

    long long s, d;
    if (wave < E) { s = ei[wave]; d = ei[E + wave]; }
    else          { s = d = (long long)(wave - E); }   // self loop

    const int H = HEADS * CH;
    const float* xls = xl + (size_t)s * H;
    const float* xrd = xr + (size_t)d * H;

#pragma unroll
    for (int h = 0; h < HEADS; ++h) {
        float part = 0.f;
        for (int c = lane; c < CH; c += 32) {
            float v = xls[h * CH + c] + xrd[h * CH + c];
            v = (v > 0.f) ? v : NEG_SLOPE * v;
            part += att[h * CH + c] * v;
        }
        float logit = wave_reduce_sum(part);
        if (lane == 0) {
            logits[(size_t)wave * HEADS + h] = logit;
            atomicMax(&mkey[(size_t)d * HEADS + h], f32_ordered_key(logit));
        }
    }
}

// ---------------------------------------------------------------------------
// Pass 2: p = exp(logit - max[dst]); segment sum.  Thread per (edge, head).
// ---------------------------------------------------------------------------
__global__ void edge_softmax_num(const long long* __restrict__ ei, int E, int Etot,
                                 float* __restrict__ logits,
                                 const unsigned int* __restrict__ mkey,
                                 float* __restrict__ ssum)
{
    int t = blockIdx.x * blockDim.x + threadIdx.x;
    if (t >= Etot * HEADS) return;
    int e = t >> 2;
    int h = t & (HEADS - 1);
    long long d = (e < E) ? ei[E + e] : (long long)(e - E);
    float m = f32_from_key(mkey[(size_t)d * HEADS + h]);
    float p = __expf(logits[t] - m);
    logits[t] = p;
    atomicAdd(&ssum[(size_t)d * HEADS + h], p);
}

// ---------------------------------------------------------------------------
// Pass 3: out[dst] += (p/s[dst]) * xl[src].  Wave per edge, lanes over channels.
// ---------------------------------------------------------------------------
__global__ void edge_scatter(const long long* __restrict__ ei, int E, int Etot,
                             const float* __restrict__ xl,
                             const float* __restrict__ p,     // [Etot, HEADS]
                             const float* __restrict__ ssum,  // [N, HEADS]
                             float* __restrict__ out,         // [N, HEADS*CH]
                             int CH)
{
    const int wave = (blockIdx.x * blockDim.x + threadIdx.x) >> 5;
    const int lane = threadIdx.x & 31;
    if (wave >= Etot) return;

    long long s, d;
    if (wave < E) { s = ei[wave]; d = ei[E + wave]; }
    else          { s = d = (long long)(wave - E); }

    const int H = HEADS * CH;
    const float* xls = xl + (size_t)s * H;
    float* od = out + (size_t)d * H;

#pragma unroll
    for (int h = 0; h < HEADS; ++h) {
        float alpha = p[(size_t)wave * HEADS + h] / ssum[(size_t)d * HEADS + h];
        for (int c = lane; c < CH; c += 32)
            atomicAdd(&od[h * CH + c], alpha * xls[h * CH + c]);
    }
}

// ---------------------------------------------------------------------------
// h = elu(acc + bias) + identity   (elementwise; hout may alias acc or id)
// ---------------------------------------------------------------------------
__global__ void finalize_elu_add(const float* acc, const float* __restrict__ bias,
                                 const float* id, float* hout, int n, int H)
{
    int t = blockIdx.x * blockDim.x + threadIdx.x;
    if (t >= n) return;
    float v = acc[t] + bias[t % H];
    float e = (v > 0.f) ? v : (__expf(v) - 1.f);
    hout[t] = e + id[t];
}

// ---------------------------------------------------------------------------
// Launcher
// ---------------------------------------------------------------------------
extern "C" void kernel_launch(void* const* d_in, const int* in_sizes, int n_in,
                              void* d_out, int out_size, void* d_ws, size_t ws_size,
                              hipStream_t stream)
{
    const float*     x     = (const float*)d_in[0];
    const long long* ei    = (const long long*)d_in[1];
    const float* Wl1  = (const float*)d_in[2];
    const float* bl1  = (const float*)d_in[3];
    const float* Wr1  = (const float*)d_in[4];
    const float* br1  = (const float*)d_in[5];
    const float* att1 = (const float*)d_in[6];
    const float* bias1= (const float*)d_in[7];
    const float* P1   = (const float*)d_in[8];
    const float* pb1  = (const float*)d_in[9];
    const float* Wl2  = (const float*)d_in[10];
    const float* bl2  = (const float*)d_in[11];
    const float* Wr2  = (const float*)d_in[12];
    const float* br2  = (const float*)d_in[13];
    const float* att2 = (const float*)d_in[14];
    const float* bias2= (const float*)d_in[15];
    const float* P2   = (const float*)d_in[16];
    const float* pb2  = (const float*)d_in[17];
    const float* Wf   = (const float*)d_in[18];
    const float* bf   = (const float*)d_in[19];
    float* out = (float*)d_out;

    const int IN_CH = 128, H1 = 256, H2 = 128, OUT_CH = 64, C1 = 64, C2 = 32;
    const int N    = in_sizes[0] / IN_CH;   // 20000 (multiple of 16)
    const int E    = in_sizes[1] / 2;       // 320000
    const int Etot = E + N;                 // 340000

    // ---- workspace layout (fp32), buffers reused across layers ----
    float* F = (float*)d_ws;
    const size_t NB = (size_t)N * H1;       // 5.12M floats
    float* bufA   = F;                      // xl1   | {xl2, id2}
    float* bufB   = F + NB;                 // xr1 -> acc1 | {acc2, xr2}
    float* bufC   = F + 2 * NB;             // id1 -> h1 (layer-2 input)
    float* logits = F + 3 * NB;             // [Etot, HEADS]
    float* ssum   = logits + (size_t)Etot * HEADS;
    unsigned int* mkey = (unsigned int*)(ssum + (size_t)N * HEADS);

    const unsigned int NEGINF_KEY = 0x007FFFFFu; // ordered key of -inf
    const int nh = N * HEADS;
    dim3 blk128(128), blk256(256);
    const int fillHB     = (nh + 255) / 256;
    const int edgeBlocks = (Etot + 7) / 8;               // 8 waves / block
    const int e2Blocks   = (Etot * HEADS + 255) / 256;

    // ================= layer 1 =================
    fill_u32_kernel<<<fillHB, blk256, 0, stream>>>(mkey, NEGINF_KEY, nh);
    fill_u32_kernel<<<fillHB, blk256, 0, stream>>>((unsigned int*)ssum, 0u, nh);

    dim3 g1(N / 16, H1 / 64);
    wmma_gemm_bias<<<g1, blk128, 0, stream>>>(x, Wl1, bl1, bufA, IN_CH, H1); // xl1
    wmma_gemm_bias<<<g1, blk128, 0, stream>>>(x, Wr1, br1, bufB, IN_CH, H1); // xr1
    wmma_gemm_bias<<<g1, blk128, 0, stream>>>(x, P1,  pb1, bufC, IN_CH, H1); // identity1

    edge_logits_max<<<edgeBlocks, blk256, 0, stream>>>(ei, E, Etot, bufA, bufB,
                                                       att1, logits, mkey, C1);
    edge_softmax_num<<<e2Blocks, blk256, 0, stream>>>(ei, E, Etot, logits, mkey, ssum);

    fill_u32_kernel<<<(int)((NB + 255) / 256), blk256, 0, stream>>>(
        (unsigned int*)bufB, 0u, (int)NB);  // xr1 dead -> reuse as accumulator
    edge_scatter<<<edgeBlocks, blk256, 0, stream>>>(ei, E, Etot, bufA, logits, ssum,
                                                    bufB, C1);
    finalize_elu_add<<<(int)((NB + 255) / 256), blk256, 0, stream>>>(
        bufB, bias1, bufC, bufC, (int)NB, H1);          // h1 -> bufC (in place)

    // ================= layer 2 =================
    float* xl2  = bufA;
    float* id2  = bufA + (size_t)N * H2;
    float* acc2 = bufB;
    float* xr2  = bufB + (size_t)N * H2;
    const size_t NB2 = (size_t)N * H2;

    fill_u32_kernel<<<fillHB, blk256, 0, stream>>>(mkey, NEGINF_KEY, nh);
    fill_u32_kernel<<<fillHB, blk256, 0, stream>>>((unsigned int*)ssum, 0u, nh);

    dim3 g2(N / 16, H2 / 64);
    wmma_gemm_bias<<<g2, blk128, 0, stream>>>(bufC, Wl2, bl2, xl2, H1, H2);
    wmma_gemm_bias<<<g2, blk128, 0, stream>>>(bufC, Wr2, br2, xr2, H1, H2);
    wmma_gemm_bias<<<g2, blk128, 0, stream>>>(bufC, P2,  pb2, id2, H1, H2);

    edge_logits_max<<<edgeBlocks, blk256, 0, stream>>>(ei, E, Etot, xl2, xr2,
                                                       att2, logits, mkey, C2);
    edge_softmax_num<<<e2Blocks, blk256, 0, stream>>>(ei, E, Etot, logits, mkey, ssum);

    fill_u32_kernel<<<(int)((NB2 + 255) / 256), blk256, 0, stream>>>(
        (unsigned int*)acc2, 0u, (int)NB2);
    edge_scatter<<<edgeBlocks, blk256, 0, stream>>>(ei, E, Etot, xl2, logits, ssum,
                                                    acc2, C2);
    finalize_elu_add<<<(int)((NB2 + 255) / 256), blk256, 0, stream>>>(
        acc2, bias2, id2, id2, (int)NB2, H2);           // h2 -> id2 (in place)

    // ================= fc =================
    dim3 g3(N / 16, OUT_CH / 64);
    wmma_gemm_bias<<<g3, blk128, 0, stream>>>(id2, Wf, bf, out, H2, OUT_CH);
}